// PillarQueryAndGroup_84267258347997
// MI455X (gfx1250) — compile-verified
//
#include <hip/hip_runtime.h>
#include <cstdint>
#include <cstddef>

// ---------------- problem constants (from reference) ----------------
constexpr int   Hh   = 1440, Ww = 1440, GZz = 32, GTt = 10, Bb = 2;
constexpr float Pf   = 0.075f, VZf = 0.25f, VTf = 0.05f;
constexpr float ZC   = 4.0f;        // (8+0)/2
constexpr float TZC  = 5.0f;
constexpr int   HW   = Hh * Ww;           // 2,073,600
constexpr int   WGZ  = Ww * GZz;          // 46,080
constexpr int   HWGZ = HW * GZz;          // 66,355,200
constexpr int   WGT  = Ww * GTt;          // 14,400
constexpr int   HWGT = HW * GTt;          // 20,736,000
// bitmap word counts per key domain (B * domain bits / 32)
constexpr int   W2 = (Bb * HW   + 31) / 32;   // 129,600
constexpr int   W3 = (Bb * HWGZ + 31) / 32;   // 4,147,200
constexpr int   WT = (Bb * HWGT + 31) / 32;   // 1,296,000

typedef unsigned int u32x4 __attribute__((ext_vector_type(4)));
typedef int          i32x8 __attribute__((ext_vector_type(8)));
typedef int          i32x4 __attribute__((ext_vector_type(4)));

// ---------------- small utility kernels ----------------
__global__ void fill_u32(unsigned* __restrict__ p, unsigned v, int n) {
  int i = blockIdx.x * blockDim.x + threadIdx.x;
  if (i < n) p[i] = v;
}

// per-point: compute 3 keys, save them, set presence bits
__global__ void keys_mark(const float* __restrict__ xyzt, const int* __restrict__ bc,
                          int* __restrict__ k2o, int* __restrict__ k3o, int* __restrict__ kto,
                          unsigned* __restrict__ bm2, unsigned* __restrict__ bm3,
                          unsigned* __restrict__ bmt, int n) {
  int i = blockIdx.x * blockDim.x + threadIdx.x;
  if (i >= n) return;
  float x = xyzt[i * 5 + 0], y = xyzt[i * 5 + 1];
  float z = xyzt[i * 5 + 2], t = xyzt[i * 5 + 4];
  int b  = (i >= bc[0]) ? 1 : 0;                 // B == 2
  int ix = (int)(x / Pf),  iy = (int)(y / Pf);
  int iz = (int)(z / VZf), it = (int)(t / VTf);
  int k2 = (b * Hh + iy) * Ww + ix;
  int k3 = k2 * GZz + iz;
  int kt = k2 * GTt + it;
  k2o[i] = k2; k3o[i] = k3; kto[i] = kt;
  atomicOr(&bm2[k2 >> 5], 1u << (k2 & 31));
  atomicOr(&bm3[k3 >> 5], 1u << (k3 & 31));
  atomicOr(&bmt[kt >> 5], 1u << (kt & 31));
}

// stage 1: per-word popcount + intra-block exclusive scan; block totals out
__global__ void scan_words(const unsigned* __restrict__ bitmap,
                           unsigned* __restrict__ prefix,
                           unsigned* __restrict__ bsum, int L) {
  __shared__ unsigned sh[256];
  int t = threadIdx.x;
  int i = blockIdx.x * 256 + t;
  unsigned v = (i < L) ? (unsigned)__popc(bitmap[i]) : 0u;
  sh[t] = v; __syncthreads();
  for (int off = 1; off < 256; off <<= 1) {
    unsigned a = (t >= off) ? sh[t - off] : 0u;
    __syncthreads();
    sh[t] += a;
    __syncthreads();
  }
  if (i < L) prefix[i] = sh[t] - v;          // exclusive within block
  if (t == 255) bsum[blockIdx.x] = sh[255];  // block total
}

// stage 2: single-block exclusive scan of block sums (L up to ~16k)
__global__ void scan_block_sums(unsigned* __restrict__ bs, int L) {
  __shared__ unsigned sh[1024];
  __shared__ unsigned carry;
  int t = threadIdx.x;
  if (t == 0) carry = 0u;
  __syncthreads();
  for (int base = 0; base < L; base += 1024) {
    int i = base + t;
    unsigned v = (i < L) ? bs[i] : 0u;
    sh[t] = v; __syncthreads();
    for (int off = 1; off < 1024; off <<= 1) {
      unsigned a = (t >= off) ? sh[t - off] : 0u;
      __syncthreads();
      sh[t] += a;
      __syncthreads();
    }
    unsigned incl  = sh[t];
    unsigned total = sh[1023];
    if (i < L) bs[i] = carry + incl - v;     // exclusive global
    __syncthreads();
    if (t == 0) carry += total;
    __syncthreads();
  }
}

// stage 3: add scanned block offsets
__global__ void scan_add(unsigned* __restrict__ prefix, const unsigned* __restrict__ bs, int L) {
  int i = blockIdx.x * 256 + threadIdx.x;
  if (i < L) prefix[i] += bs[blockIdx.x];
}

// scatter-decode each set bit at its rank. mode: 0 = 2D pillar (3 ints),
// 1 = Z grid (4 ints), 2 = T grid (4 ints)
__global__ void decode_bins(const unsigned* __restrict__ bitmap,
                            const unsigned* __restrict__ prefix,
                            int L, int* __restrict__ coords, int mode) {
  int w = blockIdx.x * blockDim.x + threadIdx.x;
  if (w >= L) return;
  unsigned bits = bitmap[w];
  unsigned rank = prefix[w];
  while (bits) {
    int b   = __ffs(bits) - 1;
    int key = w * 32 + b;
    if (mode == 0) {
      int pb = key / HW, rem = key % HW;
      coords[rank * 3 + 0] = pb;
      coords[rank * 3 + 1] = rem / Ww;
      coords[rank * 3 + 2] = rem % Ww;
    } else if (mode == 1) {
      int vb = key / HWGZ, r = key % HWGZ;
      int vy = r / WGZ;   r = r % WGZ;
      coords[rank * 4 + 0] = vb;
      coords[rank * 4 + 1] = r % GZz;   // vz
      coords[rank * 4 + 2] = vy;
      coords[rank * 4 + 3] = r / GZz;   // vx
    } else {
      int tb = key / HWGT, r = key % HWGT;
      int ty = r / WGT;   r = r % WGT;
      coords[rank * 4 + 0] = tb;
      coords[rank * 4 + 1] = r % GTt;   // tt
      coords[rank * 4 + 2] = ty;
      coords[rank * 4 + 3] = r / GTt;   // tx
    }
    rank++;
    bits &= bits - 1;
  }
}

// inverse index = rank of key = prefix popcount below the key's bit
__global__ void write_inv(const int* __restrict__ keys, const unsigned* __restrict__ bm,
                          const unsigned* __restrict__ pre, int* __restrict__ inv, int n) {
  int i = blockIdx.x * blockDim.x + threadIdx.x;
  if (i >= n) return;
  int k = keys[i];
  unsigned word = bm[k >> 5];
  unsigned mask = (1u << (k & 31)) - 1u;
  inv[i] = (int)(pre[k >> 5] + (unsigned)__popc(word & mask));
}

// ---------------- feature fan-out kernel ----------------
// pf tile (128 pts x 64 f32 = 32KB) staged into LDS by ONE TDM descriptor
// (tensor_load_to_lds, TENSORcnt); xyzt tile staged with per-lane
// global_load_async_to_lds_b32 (ASYNCcnt). Then three contiguous 70-wide
// output streams are written from LDS: pf is read from HBM exactly once
// while 3x feature streams are written.
#define FTILE 128
__global__ __launch_bounds__(FTILE) void features_kernel(
    const float* __restrict__ xyzt, const float* __restrict__ pf,
    float* __restrict__ feat, float* __restrict__ zfeat, float* __restrict__ tfeat,
    int n) {
  __shared__ float s_pf[FTILE * 64];
  __shared__ float s_x[FTILE * 5];
  int t  = threadIdx.x;
  int p0 = blockIdx.x * FTILE;

  // ---- TDM: one D# per block moves the 32KB pf tile into LDS ----
  if (t < 32) {  // wave 0 only (TDM issues per-wave; EXEC ignored)
    unsigned long long gaddr = (unsigned long long)(uintptr_t)(pf + (size_t)p0 * 64);
    unsigned lds0 = (unsigned)(uintptr_t)&s_pf[0];          // LDS byte offset
    unsigned td0  = (unsigned)((n - p0) * 64);              // valid elems from tile start
    if (td0 > (unsigned)(FTILE * 64)) td0 = (unsigned)(FTILE * 64);
    // D# group0: count=1 | lds_addr | global_addr[56:0] | type=2
    u32x4 g0;
    g0.x = 1u;
    g0.y = lds0;
    g0.z = (unsigned)(gaddr & 0xffffffffull);
    g0.w = (unsigned)((gaddr >> 32) & 0x1ffffffull) | (2u << 30);
    // D# group1: mask=0, data_size=2 (4B), 1-D tile: tile_dim0=8192,
    // tensor_dim0=td0 so OOB reads past the buffer zero-fill on last tile.
    i32x8 g1;
    g1.s0 = (int)(2u << 16);                                 // data_size=4B
    g1.s1 = (int)((td0 & 0xffffu) << 16);                    // tensor_dim0[15:0]
    g1.s2 = (int)(((td0 >> 16) & 0xffffu) | (1u << 16));     // dim0 hi | tensor_dim1=1
    g1.s3 = (int)((unsigned)(FTILE * 64) << 16);             // dim1 hi=0 | tile_dim0=8192
    g1.s4 = 0;                                               // tile_dim1=0, tile_dim2=0
    g1.s5 = (int)td0;                                        // tensor_dim0_stride (unused, 1-D)
    g1.s6 = 0;
    g1.s7 = 0;
    i32x4 z4 = {0, 0, 0, 0};                                 // groups 2/3 unused (<=2D)
    i32x8 z8 = {0, 0, 0, 0, 0, 0, 0, 0};
    __builtin_amdgcn_tensor_load_to_lds(g0, g1, z4, z4, z8, 0);
    __builtin_amdgcn_s_wait_tensorcnt(0);
  }

  // ---- async per-lane copy of the xyzt rows (ASYNCcnt path) ----
  {
    unsigned long long xb = (unsigned long long)(uintptr_t)xyzt;
    unsigned lds_x = (unsigned)(uintptr_t)&s_x[0];
    int p = p0 + t; if (p > n - 1) p = n - 1;                // clamp: dup, harmless
#pragma unroll
    for (int j = 0; j < 5; ++j) {
      unsigned goff = (unsigned)(p * 5 + j) * 4u;
      unsigned loff = lds_x + (unsigned)(t * 5 + j) * 4u;
      asm volatile("global_load_async_to_lds_b32 %0, %1, %2"
                   :: "v"(loff), "v"(goff), "s"(xb) : "memory");
    }
    asm volatile("s_wait_asynccnt 0" ::: "memory");
  }
  __syncthreads();

  int valid = n - p0; if (valid > FTILE) valid = FTILE;
  int total = valid * 70;
  float* f  = feat  + (size_t)p0 * 70;
  float* zf = zfeat + (size_t)p0 * 70;
  float* tf = tfeat + (size_t)p0 * 70;
  for (int e = t; e < total; e += FTILE) {
    int pp = e / 70, c = e - pp * 70;
    float x  = s_x[pp * 5 + 0], y = s_x[pp * 5 + 1];
    float z  = s_x[pp * 5 + 2], tv = s_x[pp * 5 + 4];
    float vf, vt;
    if (c < 64)        { vf = s_pf[pp * 64 + c]; vt = vf; }
    else if (c == 64)  { vf = x;  vt = x; }
    else if (c == 65)  { vf = y;  vt = y; }
    else if (c == 66)  { vf = z;  vt = tv; }
    else if (c == 67)  { float cx = ((float)((int)(x / Pf)) + 0.5f) * Pf; vf = x - cx; vt = vf; }
    else if (c == 68)  { float cy = ((float)((int)(y / Pf)) + 0.5f) * Pf; vf = y - cy; vt = vf; }
    else               { vf = z - ZC; vt = tv - TZC; }
    f[e]  = vf;   // features
    zf[e] = vf;   // z_features (provably identical to features)
    tf[e] = vt;   // t_features
  }
}

// ---------------- host launch ----------------
extern "C" void kernel_launch(void* const* d_in, const int* in_sizes, int n_in,
                              void* d_out, int out_size, void* d_ws, size_t ws_size,
                              hipStream_t stream) {
  const float* xyzt = (const float*)d_in[0];
  const int*   bc   = (const int*)d_in[1];
  const float* pf   = (const float*)d_in[2];
  float* out = (float*)d_out;
  char*  ws  = (char*)d_ws;
  const int n = in_sizes[0] / 5;   // N points

  // workspace layout (bytes)
  size_t o = 0;
  int*      keys2 = (int*)(ws + o);      o += (size_t)n * 4;
  int*      keys3 = (int*)(ws + o);      o += (size_t)n * 4;
  int*      keysT = (int*)(ws + o);      o += (size_t)n * 4;
  unsigned* bm2   = (unsigned*)(ws + o); o += (size_t)W2 * 4;
  unsigned* bm3   = (unsigned*)(ws + o); o += (size_t)W3 * 4;
  unsigned* bmT   = (unsigned*)(ws + o); o += (size_t)WT * 4;
  unsigned* pre2  = (unsigned*)(ws + o); o += (size_t)W2 * 4;
  unsigned* pre3  = (unsigned*)(ws + o); o += (size_t)W3 * 4;
  unsigned* preT  = (unsigned*)(ws + o); o += (size_t)WT * 4;
  const int nb2 = (W2 + 255) / 256, nb3 = (W3 + 255) / 256, nbT = (WT + 255) / 256;
  unsigned* bs2 = (unsigned*)(ws + o);   o += (size_t)((nb2 + 1023) & ~1023) * 4;
  unsigned* bs3 = (unsigned*)(ws + o);   o += (size_t)((nb3 + 1023) & ~1023) * 4;
  unsigned* bsT = (unsigned*)(ws + o);   o += (size_t)((nbT + 1023) & ~1023) * 4;

  // output layout (elements, in return order)
  int*   oC2 = (int*)(out + 0);
  int*   oI2 = (int*)(out + (size_t)3 * n);
  float* oF  =        out + (size_t)4 * n;
  int*   oC3 = (int*)(out + (size_t)74 * n);
  int*   oI3 = (int*)(out + (size_t)78 * n);
  float* oZF =        out + (size_t)79 * n;
  int*   oCT = (int*)(out + (size_t)149 * n);
  int*   oIT = (int*)(out + (size_t)153 * n);
  float* oTF =        out + (size_t)154 * n;

  // 1) zero all three bitmaps (contiguous) + fill coord outputs with -1
  {
    int tw = W2 + W3 + WT;
    fill_u32<<<(tw + 255) / 256, 256, 0, stream>>>(bm2, 0u, tw);
    fill_u32<<<(3 * n + 255) / 256, 256, 0, stream>>>((unsigned*)oC2, 0xFFFFFFFFu, 3 * n);
    fill_u32<<<(4 * n + 255) / 256, 256, 0, stream>>>((unsigned*)oC3, 0xFFFFFFFFu, 4 * n);
    fill_u32<<<(4 * n + 255) / 256, 256, 0, stream>>>((unsigned*)oCT, 0xFFFFFFFFu, 4 * n);
  }

  // 2) keys + presence bits
  keys_mark<<<(n + 255) / 256, 256, 0, stream>>>(xyzt, bc, keys2, keys3, keysT,
                                                 bm2, bm3, bmT, n);

  // 3) hierarchical exclusive popcount-scan per bitmap
  scan_words<<<nb2, 256, 0, stream>>>(bm2, pre2, bs2, W2);
  scan_block_sums<<<1, 1024, 0, stream>>>(bs2, nb2);
  scan_add<<<nb2, 256, 0, stream>>>(pre2, bs2, W2);

  scan_words<<<nb3, 256, 0, stream>>>(bm3, pre3, bs3, W3);
  scan_block_sums<<<1, 1024, 0, stream>>>(bs3, nb3);
  scan_add<<<nb3, 256, 0, stream>>>(pre3, bs3, W3);

  scan_words<<<nbT, 256, 0, stream>>>(bmT, preT, bsT, WT);
  scan_block_sums<<<1, 1024, 0, stream>>>(bsT, nbT);
  scan_add<<<nbT, 256, 0, stream>>>(preT, bsT, WT);

  // 4) decode unique bins at their ranks
  decode_bins<<<(W2 + 255) / 256, 256, 0, stream>>>(bm2, pre2, W2, oC2, 0);
  decode_bins<<<(W3 + 255) / 256, 256, 0, stream>>>(bm3, pre3, W3, oC3, 1);
  decode_bins<<<(WT + 255) / 256, 256, 0, stream>>>(bmT, preT, WT, oCT, 2);

  // 5) inverse indices
  write_inv<<<(n + 255) / 256, 256, 0, stream>>>(keys2, bm2, pre2, oI2, n);
  write_inv<<<(n + 255) / 256, 256, 0, stream>>>(keys3, bm3, pre3, oI3, n);
  write_inv<<<(n + 255) / 256, 256, 0, stream>>>(keysT, bmT, preT, oIT, n);

  // 6) feature fan-out (pf read once via TDM->LDS, 3 streams written)
  features_kernel<<<(n + FTILE - 1) / FTILE, FTILE, 0, stream>>>(xyzt, pf, oF, oZF, oTF, n);

  (void)n_in; (void)out_size; (void)ws_size;
}